// Gemma3nAudioRelativePositionEmbedding_704374636652
// MI455X (gfx1250) — compile-verified
//
#include <hip/hip_runtime.h>
#include <hip/hip_bf16.h>

// ---------------------------------------------------------------------------
// Gemma3n audio relative-position embedding, MI455X (gfx1250).
//
// Roofline: ~472 MB moved, ~2.8 GFLOP -> bandwidth-bound (~20us @ 23.3 TB/s).
// Strategy: single streaming pass over Q/K, contiguous float4 staging into
// LDS (300 KB/WGP, within the 320 KB budget), f32 WMMA 16x16x4 for the
// contraction, relative-shift fused into the epilogue.
// ---------------------------------------------------------------------------

typedef float v2f __attribute__((ext_vector_type(2)));
typedef float v8f __attribute__((ext_vector_type(8)));

#define NUM_HEADS 8
#define HEAD_DIM  192
#define W_DIM     12      // query window
#define C_DIM     24      // key context
#define F_DIM     13      // relative positions
#define PITCH     196     // LDS row pitch in floats; 196 % 64 == 4 -> conflict-free frags
#define HEAD_ROWS (W_DIM + C_DIM + F_DIM)         // 12 Q rows + 37 B-source rows
#define HEAD_LDS  (HEAD_ROWS * PITCH)             // 9604 floats per head
#define LDS_FLOATS (NUM_HEADS * HEAD_LDS)         // 76832 floats
#define LDS_BYTES  (LDS_FLOATS * 4)               // 307,328 B  (< 320 KB WGP limit)

// ---------------------------------------------------------------------------
// Kernel 1: E[f][n*192+h] = sum_t sin_emb[f][t] * W[t][n*192+h]
// 13 x 1536 output, K=1536. Tiny (61 MFLOP, ~13x9.4MB mostly-L2 reads).
// grid = (2, 13), block = 256; each thread produces 3 columns.
// ---------------------------------------------------------------------------
__global__ void __launch_bounds__(256) pos_proj_kernel(
    const float* __restrict__ Wp,   // (1536, 1536) row-major
    float* __restrict__ E)          // (13, 1536) row-major
{
    __shared__ float s[1536];
    const int f    = blockIdx.y;          // 0..12
    const int half = blockIdx.x;          // 0..1
    const int tid  = threadIdx.x;         // 0..255
    const float pos = (float)(12 - f);    // arange(12, -1, -1)
    const float lti = 9.2103403719761836f / 767.0f;  // log(10000)/(768-1)

    for (int t = tid; t < 1536; t += 256) {
        int i = (t < 768) ? t : (t - 768);
        float sc = pos * expf(-(float)i * lti);
        s[t] = (t < 768) ? sinf(sc) : cosf(sc);
    }
    __syncthreads();

    float a0 = 0.f, a1 = 0.f, a2 = 0.f;
    const int c0 = half * 768 + tid;
    for (int t = 0; t < 1536; ++t) {
        const float sv = s[t];
        const float* wr = Wp + (size_t)t * 1536 + c0;
        a0 += sv * wr[0];
        a1 += sv * wr[256];
        a2 += sv * wr[512];
    }
    float* er = E + (size_t)f * 1536 + c0;
    er[0] = a0; er[256] = a1; er[512] = a2;
}

// ---------------------------------------------------------------------------
// Kernel 2: main. One block per (b,u); 8 waves, one head per wave.
// D = Q(12x192) x [K | E]^T (192x37) via V_WMMA_F32_16X16X4_F32, 3 N-tiles,
// 48 K-chunks. Epilogue applies relative shift:
//   out[w][c] = D[w][c] + (0 <= c-w <= 12 ? D[w][24 + c-w] : 0)
// ---------------------------------------------------------------------------
__global__ void __launch_bounds__(256) relpos_attn_kernel(
    const float* __restrict__ Qg_,   // (16,128,12,8,192)
    const float* __restrict__ Kg_,   // (16,128,24,8,192)
    const float* __restrict__ Eg,    // (13,1536) from d_ws
    float* __restrict__ out)         // (16,8,128,12,24)
{
    extern __shared__ float lds[];
    const int tid = threadIdx.x;
    const int bu  = blockIdx.x;                 // b*128 + u
    const float* Qg = Qg_ + (size_t)bu * (W_DIM * NUM_HEADS * HEAD_DIM);   // 18432
    const float* Kg = Kg_ + (size_t)bu * (C_DIM * NUM_HEADS * HEAD_DIM);   // 36864

    // ---- stage Q slab (contiguous 72 KB) -> per-head padded rows --------
    for (int k = tid; k < 4608; k += 256) {          // 18432/4 float4s
        float4 v = ((const float4*)Qg)[k];
        int e = k << 2;
        int h = e % 192;
        int n = (e / 192) & 7;
        int w = e / 1536;
        *(float4*)&lds[n * HEAD_LDS + w * PITCH + h] = v;
    }
    // ---- stage K slab (contiguous 144 KB) -> B-source rows 0..23 --------
    for (int k = tid; k < 9216; k += 256) {          // 36864/4 float4s
        float4 v = ((const float4*)Kg)[k];
        int e = k << 2;
        int h = e % 192;
        int n = (e / 192) & 7;
        int c = e / 1536;
        *(float4*)&lds[n * HEAD_LDS + (W_DIM + c) * PITCH + h] = v;
    }
    // ---- stage E (13x1536, L2-resident) -> B-source rows 24..36 ---------
    for (int k = tid; k < 4992; k += 256) {          // 19968/4 float4s
        float4 v = ((const float4*)Eg)[k];
        int e = k << 2;
        int f = e / 1536;
        int r = e % 1536;
        int n = r / 192;
        int h = r % 192;
        *(float4*)&lds[n * HEAD_LDS + (W_DIM + C_DIM + f) * PITCH + h] = v;
    }
    __syncthreads();

    // ---- WMMA compute: wave wv owns head wv ------------------------------
    const int wv   = tid >> 5;         // head index (wave32)
    const int lane = tid & 31;
    const int l16  = lane & 15;
    const int hi   = lane >> 4;        // 0: K=+0,+1   1: K=+2,+3 (A & B frags)

    const float* hb = &lds[wv * HEAD_LDS];
    // A frag: lane row = l16 (clamp 12..15 -> 11, results ignored; broadcast read)
    const float* qb = hb + min(l16, 11) * PITCH + 2 * hi;
    // B frags: source row = tile*16 + l16 (tile2 clamps to row 36)
    const float* b0 = hb + (W_DIM + l16) * PITCH + 2 * hi;
    const float* b1 = hb + (W_DIM + 16 + l16) * PITCH + 2 * hi;
    const float* b2 = hb + (W_DIM + min(32 + l16, 36)) * PITCH + 2 * hi;

    v8f acc0 = {0.f,0.f,0.f,0.f,0.f,0.f,0.f,0.f};
    v8f acc1 = {0.f,0.f,0.f,0.f,0.f,0.f,0.f,0.f};
    v8f acc2 = {0.f,0.f,0.f,0.f,0.f,0.f,0.f,0.f};

#pragma unroll 8
    for (int kk = 0; kk < HEAD_DIM; kk += 4) {
        v2f av  = *(const v2f*)(qb + kk);
        v2f bv0 = *(const v2f*)(b0 + kk);
        v2f bv1 = *(const v2f*)(b1 + kk);
        v2f bv2 = *(const v2f*)(b2 + kk);
        acc0 = __builtin_amdgcn_wmma_f32_16x16x4_f32(false, av, false, bv0,
                                                     (short)0, acc0, false, false);
        acc1 = __builtin_amdgcn_wmma_f32_16x16x4_f32(false, av, false, bv1,
                                                     (short)0, acc1, false, false);
        acc2 = __builtin_amdgcn_wmma_f32_16x16x4_f32(false, av, false, bv2,
                                                     (short)0, acc2, false, false);
    }

    // ---- epilogue: dump D to (dead) LDS, fuse relative shift ------------
    __syncthreads();   // all waves done reading staged data before overwrite

    float* dbase = &lds[wv * (16 * 49)];   // 16 rows x 48 cols, pitch 49
#pragma unroll
    for (int r = 0; r < 8; ++r) {
        const int m = r + 8 * hi;          // C/D layout: VGPR r holds rows r, r+8
        dbase[m * 49 + l16 +  0] = acc0[r];
        dbase[m * 49 + l16 + 16] = acc1[r];
        dbase[m * 49 + l16 + 32] = acc2[r];
    }
    // same-wave LDS RAW: DS ops are in-order within a wave; compiler inserts waits

    const int b = bu >> 7, u = bu & 127;
    float* op = out + ((((size_t)b * NUM_HEADS + wv) * 128 + u) * (W_DIM * C_DIM));
    for (int e = lane; e < W_DIM * C_DIM; e += 32) {
        const int w = e / C_DIM, c = e % C_DIM;
        float v = dbase[w * 49 + c];               // term_ac
        const int f = c - w;                       // shifted bd index
        if (f >= 0 && f < F_DIM) v += dbase[w * 49 + C_DIM + f];
        op[e] = v;
    }
}

// ---------------------------------------------------------------------------
extern "C" void kernel_launch(void* const* d_in, const int* in_sizes, int n_in,
                              void* d_out, int out_size, void* d_ws, size_t ws_size,
                              hipStream_t stream) {
    const float* Q  = (const float*)d_in[0];   // (16,128,12,8,192) f32
    const float* K  = (const float*)d_in[1];   // (16,128,24,8,192) f32
    const float* Wp = (const float*)d_in[2];   // (1536,1536) f32
    float* out = (float*)d_out;                // (16,8,128,12,24) f32
    float* E   = (float*)d_ws;                 // 13*1536 f32 scratch

    pos_proj_kernel<<<dim3(2, 13), 256, 0, stream>>>(Wp, E);
    relpos_attn_kernel<<<16 * 128, 256, LDS_BYTES, stream>>>(Q, K, E, out);
}